// SharedGNNBackbone_68358699483269
// MI455X (gfx1250) — compile-verified
//
#include <hip/hip_runtime.h>
#include <math.h>

// ---------------- problem constants (match reference) ----------------
#define N_NODES 100000
#define N_EDGES 1600000
#define E_TOT   (N_EDGES + N_NODES)   // with self loops
#define HC      128                   // heads*channels, same every layer
#define H_MAX   4

typedef float v2f __attribute__((ext_vector_type(2)));
typedef float v8f __attribute__((ext_vector_type(8)));

// order-preserving float <-> int key for atomicMax-based segment max
__device__ __forceinline__ int enc_f32(float f) {
    int i = __float_as_int(f);
    return (i >= 0) ? i : (i ^ 0x7FFFFFFF);
}
__device__ __forceinline__ float dec_f32(int k) {
    int i = (k >= 0) ? k : (k ^ 0x7FFFFFFF);
    return __int_as_float(i);
}

// ---------------- per-layer scratch init ----------------
__global__ void init_kernel(float* __restrict__ agg, int* __restrict__ m_int,
                            float* __restrict__ denom, float* __restrict__ sums,
                            float* __restrict__ sumsq) {
    size_t i = (size_t)blockIdx.x * blockDim.x + threadIdx.x;
    if (i < (size_t)N_NODES * HC) agg[i] = 0.f;
    if (i < (size_t)N_NODES * H_MAX) { m_int[i] = (int)0x80000000; denom[i] = 0.f; }
    if (i < HC) { sums[i] = 0.f; sumsq[i] = 0.f; }
}

// ---------------- dense GEMM  C[N,128] = A[N,K] @ B[K,128]  via V_WMMA_F32_16X16X4_F32 ----
// block = 256 threads = 8 waves; block covers 16 rows x 128 cols; wave w -> cols [16w,16w+16)
// fp32 WMMA fragment layout (ISA 7.12.2):
//   A 16x4 : lanes 0-15 -> {K=0,K=1} in v0,v1 ; lanes 16-31 -> {K=2,K=3}
//   B 4x16 : v0: lanes0-15 row K=0 / lanes16-31 row K=2 ; v1: K=1 / K=3
//   C 16x16: vgpr i: lanes0-15 -> M=i ; lanes16-31 -> M=i+8
__global__ __launch_bounds__(256) void gemm_wmma_kernel(const float* __restrict__ A,
                                                        const float* __restrict__ B,
                                                        float* __restrict__ C, int K) {
    const int lane = threadIdx.x & 31;
    const int wave = threadIdx.x >> 5;
    const int row0 = blockIdx.x * 16;
    const int n0   = wave * 16;
    const int l15  = lane & 15;
    const int kh   = lane >> 4;      // 0 or 1 (K half)

    v8f acc = {0.f, 0.f, 0.f, 0.f, 0.f, 0.f, 0.f, 0.f};
    const float* arow = A + (size_t)(row0 + l15) * K + kh * 2;
    const float* bcol = B + n0 + l15;

    for (int k0 = 0; k0 < K; k0 += 4) {
        v2f a, b;
        a.x = arow[k0];
        a.y = arow[k0 + 1];
        b.x = bcol[(size_t)(k0 + kh * 2)     * HC];
        b.y = bcol[(size_t)(k0 + kh * 2 + 1) * HC];
        acc = __builtin_amdgcn_wmma_f32_16x16x4_f32(false, a, false, b,
                                                    (short)0, acc, false, false);
    }
    float* crow = C + (size_t)(row0 + kh * 8) * HC + n0 + l15;
#pragma unroll
    for (int i = 0; i < 8; ++i) crow[(size_t)i * HC] = acc[i];
}

// ---------------- per-node attention coefficients ----------------
// alpha_src[n,h] = sum_c h[n,h*C+c] * a_src[h*C+c]   (a_* flat over 128)
__global__ void alpha_kernel(const float* __restrict__ h, const float* __restrict__ asrc,
                             const float* __restrict__ adst, float* __restrict__ als,
                             float* __restrict__ ald, int H, int C) {
    int n = blockIdx.x * blockDim.x + threadIdx.x;
    if (n >= N_NODES) return;
    float ss[H_MAX] = {0.f, 0.f, 0.f, 0.f};
    float dd[H_MAX] = {0.f, 0.f, 0.f, 0.f};
    const float* row = h + (size_t)n * HC;
    for (int c = 0; c < HC; ++c) {
        float v = row[c];
        int hd = c / C;
        ss[hd] += v * asrc[c];
        dd[hd] += v * adst[c];
    }
    for (int hd = 0; hd < H; ++hd) {
        als[n * H + hd] = ss[hd];
        ald[n * H + hd] = dd[hd];
    }
}

// ---------------- edge pass 1: leaky-relu logits + segment max ----------------
__global__ void edge_logit_kernel(const int* __restrict__ ei, const float* __restrict__ als,
                                  const float* __restrict__ ald, float* __restrict__ ebuf,
                                  int* __restrict__ m_int, int H) {
    int t = blockIdx.x * blockDim.x + threadIdx.x;
    if (t >= E_TOT * H) return;
    int e = t / H, hd = t - e * H;
    int s, d;
    if (e < N_EDGES) { s = ei[e]; d = ei[N_EDGES + e]; }
    else             { s = d = e - N_EDGES; }          // self loop
    float v = als[s * H + hd] + ald[d * H + hd];
    v = (v > 0.f) ? v : 0.2f * v;                      // leaky relu, slope 0.2
    ebuf[t] = v;
    atomicMax(&m_int[d * H + hd], enc_f32(v));
}

// ---------------- edge pass 2: exp(e - max) + segment sum ----------------
__global__ void edge_exp_kernel(const int* __restrict__ ei, float* __restrict__ ebuf,
                                const int* __restrict__ m_int, float* __restrict__ denom,
                                int H) {
    int t = blockIdx.x * blockDim.x + threadIdx.x;
    if (t >= E_TOT * H) return;
    int e = t / H, hd = t - e * H;
    int d = (e < N_EDGES) ? ei[N_EDGES + e] : (e - N_EDGES);
    float mx = dec_f32(m_int[d * H + hd]);             // finite: every node has a self loop
    float ex = expf(ebuf[t] - mx);
    ebuf[t] = ex;
    atomicAdd(&denom[d * H + hd], ex);
}

// ---------------- edge pass 3: weighted scatter-add of features ----------------
// 32 lanes per edge, 4 channels per lane (128 channels)
__global__ void edge_agg_kernel(const int* __restrict__ ei, const float* __restrict__ h,
                                const float* __restrict__ ebuf, const float* __restrict__ denom,
                                float* __restrict__ agg, int H, int C) {
    long long t = (long long)blockIdx.x * blockDim.x + threadIdx.x;
    if (t >= (long long)E_TOT * 32) return;
    int e    = (int)(t >> 5);
    int lane = (int)(t & 31);
    int s, d;
    if (e < N_EDGES) { s = ei[e]; d = ei[N_EDGES + e]; }
    else             { s = d = e - N_EDGES; }
    int c0 = lane * 4;
    int hd = c0 / C;
    float w = ebuf[e * H + hd] / (denom[d * H + hd] + 1e-16f);
    const float4 hv = *(const float4*)(h + (size_t)s * HC + c0);
    float* ao = agg + (size_t)d * HC + c0;
    atomicAdd(ao + 0, hv.x * w);
    atomicAdd(ao + 1, hv.y * w);
    atomicAdd(ao + 2, hv.z * w);
    atomicAdd(ao + 3, hv.w * w);
}

// ---------------- BN statistics (sum, sumsq per channel) ----------------
#define STAT_ROWS 80   // 100000 / 80 = 1250 blocks exactly
__global__ __launch_bounds__(256) void bn_stats_kernel(const float* __restrict__ x,
                                                       float* __restrict__ sums,
                                                       float* __restrict__ sumsq) {
    __shared__ float ls[256], lq[256];
    int c    = threadIdx.x & 127;
    int half = threadIdx.x >> 7;
    int r0   = blockIdx.x * STAT_ROWS;
    float s = 0.f, q = 0.f;
    for (int r = r0 + half; r < r0 + STAT_ROWS; r += 2) {
        float v = x[(size_t)r * HC + c];
        s += v; q += v * v;
    }
    ls[threadIdx.x] = s; lq[threadIdx.x] = q;
    __syncthreads();
    if (half == 0) {
        atomicAdd(&sums[c],  ls[c] + ls[c + 128]);
        atomicAdd(&sumsq[c], lq[c] + lq[c + 128]);
    }
}

// ---------------- BN normalize + ReLU ----------------
// NOTE: the GAT bias b is folded away exactly: BN subtracts the per-channel mean,
// which shifts by exactly b, so (x+b) - (mu+b) == x - mu and var is unchanged.
__global__ void bn_relu_kernel(const float* __restrict__ x, const float* __restrict__ sums,
                               const float* __restrict__ sumsq, const float* __restrict__ g,
                               const float* __restrict__ be, float* __restrict__ out) {
    size_t t = (size_t)blockIdx.x * blockDim.x + threadIdx.x;
    if (t >= (size_t)N_NODES * HC) return;
    int c = (int)(t & (HC - 1));
    const float invN = 1.f / (float)N_NODES;
    float mean = sums[c] * invN;
    float var  = sumsq[c] * invN - mean * mean;
    float y = g[c] * (x[t] - mean) * rsqrtf(var + 1e-5f) + be[c];
    out[t] = (y > 0.f) ? y : 0.f;
}

// ---------------- per-layer driver ----------------
static void run_layer(const int* ei, const float* in, int K, int H, int C,
                      const float* W, const float* asrc, const float* adst,
                      const float* g, const float* be,
                      float* bufH, float* bufAgg, float* out,
                      float* als, float* ald, int* m_int, float* denom,
                      float* ebuf, float* sums, float* sumsq, hipStream_t stream) {
    const size_t tot = (size_t)N_NODES * HC;
    init_kernel<<<(int)((tot + 255) / 256), 256, 0, stream>>>(bufAgg, m_int, denom, sums, sumsq);
    gemm_wmma_kernel<<<N_NODES / 16, 256, 0, stream>>>(in, W, bufH, K);
    alpha_kernel<<<(N_NODES + 255) / 256, 256, 0, stream>>>(bufH, asrc, adst, als, ald, H, C);
    int totEH = E_TOT * H;
    edge_logit_kernel<<<(totEH + 255) / 256, 256, 0, stream>>>(ei, als, ald, ebuf, m_int, H);
    edge_exp_kernel<<<(totEH + 255) / 256, 256, 0, stream>>>(ei, ebuf, m_int, denom, H);
    long long totE32 = (long long)E_TOT * 32;
    edge_agg_kernel<<<(int)((totE32 + 255) / 256), 256, 0, stream>>>(ei, bufH, ebuf, denom,
                                                                    bufAgg, H, C);
    bn_stats_kernel<<<N_NODES / STAT_ROWS, 256, 0, stream>>>(bufAgg, sums, sumsq);
    bn_relu_kernel<<<(int)((tot + 255) / 256), 256, 0, stream>>>(bufAgg, sums, sumsq, g, be, out);
}

extern "C" void kernel_launch(void* const* d_in, const int* in_sizes, int n_in,
                              void* d_out, int out_size, void* d_ws, size_t ws_size,
                              hipStream_t stream) {
    // setup_inputs() order
    const float* x    = (const float*)d_in[0];
    const int*   ei   = (const int*)d_in[1];     // [2, N_EDGES] (int32 under default JAX x64-off)
    const float* W1   = (const float*)d_in[2];
    const float* as1  = (const float*)d_in[3];
    const float* ad1  = (const float*)d_in[4];
    // d_in[5] = b1 (zeros; folded away by BN — see bn_relu_kernel note)
    const float* g1   = (const float*)d_in[6];
    const float* be1  = (const float*)d_in[7];
    const float* W2   = (const float*)d_in[8];
    const float* as2  = (const float*)d_in[9];
    const float* ad2  = (const float*)d_in[10];
    const float* g2   = (const float*)d_in[12];
    const float* be2  = (const float*)d_in[13];
    const float* W3   = (const float*)d_in[14];
    const float* as3  = (const float*)d_in[15];
    const float* ad3  = (const float*)d_in[16];
    const float* g3   = (const float*)d_in[18];
    const float* be3  = (const float*)d_in[19];
    (void)in_sizes; (void)n_in; (void)out_size; (void)ws_size;

    // workspace carve-out (256B aligned slices)
    char* ws = (char*)d_ws;
    auto carve = [&](size_t bytes) {
        void* p = (void*)ws;
        ws += (bytes + 255) & ~(size_t)255;
        return p;
    };
    float* bufH   = (float*)carve((size_t)N_NODES * HC * 4);      // GEMM output h
    float* bufAgg = (float*)carve((size_t)N_NODES * HC * 4);      // aggregation accumulator
    float* bufAct = (float*)carve((size_t)N_NODES * HC * 4);      // BN-ReLU activations
    float* als    = (float*)carve((size_t)N_NODES * H_MAX * 4);
    float* ald    = (float*)carve((size_t)N_NODES * H_MAX * 4);
    int*   m_int  = (int*)  carve((size_t)N_NODES * H_MAX * 4);
    float* denom  = (float*)carve((size_t)N_NODES * H_MAX * 4);
    float* ebuf   = (float*)carve((size_t)E_TOT  * H_MAX * 4);
    float* sums   = (float*)carve(HC * 4);
    float* sumsq  = (float*)carve(HC * 4);

    // layer 1: K=64,  H=4, C=32
    run_layer(ei, x,      64,  4, 32,  W1, as1, ad1, g1, be1,
              bufH, bufAgg, bufAct, als, ald, m_int, denom, ebuf, sums, sumsq, stream);
    // layer 2: K=128, H=4, C=32
    run_layer(ei, bufAct, 128, 4, 32,  W2, as2, ad2, g2, be2,
              bufH, bufAgg, bufAct, als, ald, m_int, denom, ebuf, sums, sumsq, stream);
    // layer 3: K=128, H=1, C=128 -> d_out
    run_layer(ei, bufAct, 128, 1, 128, W3, as3, ad3, g3, be3,
              bufH, bufAgg, (float*)d_out, als, ald, m_int, denom, ebuf, sums, sumsq, stream);
}